// MNN_67551245631631
// MI455X (gfx1250) — compile-verified
//
#include <hip/hip_runtime.h>

// Tropical (min/max-plus) 3x3 "hit-miss" convolution for MNN.
// WMMA is inapplicable (min/max over subtraction is not a ring op the matrix
// core supports), so the kernel is VALU-f32 bound. CDNA5-specific path used:
// async global->LDS staging (ASYNCcnt) of the input tile + LDS-tiled compute.

#define HAVE_ASYNC_LDS                                                        \
  (__has_builtin(__builtin_amdgcn_global_load_async_to_lds_b128) &&           \
   __has_builtin(__builtin_amdgcn_s_wait_asynccnt))

// Pointer types matching the builtin's signature:
//   (addrspace(1) int __vector_size__(16)*, addrspace(3) ..., Imm, Imm)
typedef int v4i_ __attribute__((__vector_size__(16)));
typedef __attribute__((address_space(1))) v4i_* gbl_v4i;
typedef __attribute__((address_space(3))) v4i_* lds_v4i;

namespace {
constexpr int B_    = 8;
constexpr int CIN   = 16;
constexpr int H_    = 64;
constexpr int W_    = 64;
constexpr int COUT  = 32;
constexpr int KS    = 3;
constexpr int NW    = CIN * KS * KS;   // 144 reduction terms
constexpr int TILE_R = 8;              // output rows per block
constexpr int TROWS  = TILE_R + 2;     // + halo
constexpr int TCOLS  = 68;             // 64 cols + pad (breaks 64-bank period)
constexpr int BLOCK  = 128;            // 4 waves (wave32)
constexpr int CGRP   = 8;              // c_outs per block (4 groups in grid.z)
constexpr float FMAXV = 3.402823466e38f;
}  // namespace

__global__ __launch_bounds__(BLOCK)
void mnn_tropical_kernel(const float* __restrict__ x,
                         const float* __restrict__ Khit,
                         const float* __restrict__ Kmiss,
                         float* __restrict__ out) {
  // 4 leading pad floats so tile[-1] is a valid, zeroed LDS slot.
  __shared__ float tileRaw[4 + CIN * TROWS * TCOLS];
  __shared__ float wlds[NW * CGRP * 2];  // [(s*CGRP+cl)*2 + {hit,miss}]
  float* tile = tileRaw + 4;

  const int tid     = threadIdx.x;
  const int rowTile = blockIdx.x;          // 0..7
  const int b       = blockIdx.y;          // 0..7
  const int cgrp    = blockIdx.z;          // 0..3
  const int row0    = rowTile * TILE_R;

  const int tx = tid & 15;                 // 16 column groups of 4 px
  const int ty = tid >> 4;                 // 8 rows
  const int w0 = tx * 4;

  // ---- Stage input tile: CIN x TROWS rows, 64 interior cols, b128 chunks ----
  for (int u = tid; u < CIN * TROWS * 16; u += BLOCK) {
    const int chunk = u & 15;
    const int rl    = u >> 4;              // 0..159
    const int cin   = rl / TROWS;
    const int r     = rl - cin * TROWS;
    const int g     = row0 - 1 + r;        // global input row (pad=1)
    float* ldst = &tile[(cin * TROWS + r) * TCOLS + chunk * 4];
    if (g >= 0 && g < H_) {
      const float* gsrc = &x[((b * CIN + cin) * H_ + g) * W_ + chunk * 4];
#if HAVE_ASYNC_LDS
      __builtin_amdgcn_global_load_async_to_lds_b128(
          (gbl_v4i)gsrc, (lds_v4i)ldst, /*imm offset*/ 0, /*cpol*/ 0);
#else
      *(float4*)ldst = *(const float4*)gsrc;
#endif
    } else {
      // zero-padded rows (top/bottom image border) — plain LDS stores,
      // disjoint from any async-load destination.
      *(float4*)ldst = make_float4(0.f, 0.f, 0.f, 0.f);
    }
  }

  // Right halo cols 64..67 := 0 (so col -1 of next row and col 64 read as the
  // zero padding), plus the 4 leading pad floats.
  if (tid == 0) {
    tileRaw[0] = 0.f; tileRaw[1] = 0.f; tileRaw[2] = 0.f; tileRaw[3] = 0.f;
  }
  for (int rl = tid; rl < CIN * TROWS; rl += BLOCK) {
    *(float4*)&tile[rl * TCOLS + 64] = make_float4(0.f, 0.f, 0.f, 0.f);
  }

  // ---- Stage this block's CGRP-c_out weight slice, hit/miss interleaved ----
  for (int u = tid; u < NW * CGRP * 2; u += BLOCK) {
    const int hm = u & 1;
    const int cl = (u >> 1) & (CGRP - 1);
    const int s  = u >> 4;                  // cin*9 + ki*3 + kj
    const int cout = cgrp * CGRP + cl;
    wlds[u] = (hm ? Kmiss : Khit)[cout * NW + s];
  }

#if HAVE_ASYNC_LDS
  __builtin_amdgcn_s_wait_asynccnt(0);      // our wave's async loads landed
#endif
  __syncthreads();                          // all waves' staging visible

  const int h = row0 + ty;

  for (int cc = 0; cc < CGRP; cc += 4) {    // 4 c_outs per accumulator chunk
    float hit[4][4], mis[4][4];
#pragma unroll
    for (int c = 0; c < 4; ++c)
#pragma unroll
      for (int p = 0; p < 4; ++p) { hit[c][p] = FMAXV; mis[c][p] = -FMAXV; }

    for (int cin = 0; cin < CIN; ++cin) {
#pragma unroll
      for (int ki = 0; ki < KS; ++ki) {
        // 6-wide input window (x cols w0-1 .. w0+4), all LDS-aligned accesses
        const int base = (cin * TROWS + ty + ki) * TCOLS + w0;
        const float4 pa  = *(const float4*)&tile[base];
        const float  pm1 = tile[base - 1];
        const float  pp4 = tile[base + 4];
        const float p6[6] = {pm1, pa.x, pa.y, pa.z, pa.w, pp4};
#pragma unroll
        for (int kj = 0; kj < KS; ++kj) {
          const int wb = ((cin * 9 + ki * 3 + kj) * CGRP + cc) * 2;
          const float4 wA = *(const float4*)&wlds[wb];      // kh0 km0 kh1 km1
          const float4 wB = *(const float4*)&wlds[wb + 4];  // kh2 km2 kh3 km3
          const float kh[4] = {wA.x, wA.z, wB.x, wB.z};
          const float km[4] = {wA.y, wA.w, wB.y, wB.w};
#pragma unroll
          for (int c = 0; c < 4; ++c) {
#pragma unroll
            for (int p = 0; p < 4; ++p) {
              const float v = p6[kj + p];
              hit[c][p] = fminf(hit[c][p], v - kh[c]);
              mis[c][p] = fmaxf(mis[c][p], v - km[c]);
            }
          }
        }
      }
    }

#pragma unroll
    for (int c = 0; c < 4; ++c) {
      const int cout = cgrp * CGRP + cc + c;
      float4 o;
      o.x = hit[c][0] - mis[c][0];
      o.y = hit[c][1] - mis[c][1];
      o.z = hit[c][2] - mis[c][2];
      o.w = hit[c][3] - mis[c][3];
      *(float4*)&out[((b * COUT + cout) * H_ + h) * W_ + w0] = o;
    }
  }
}

extern "C" void kernel_launch(void* const* d_in, const int* in_sizes, int n_in,
                              void* d_out, int out_size, void* d_ws, size_t ws_size,
                              hipStream_t stream) {
  const float* x  = (const float*)d_in[0];
  const float* kh = (const float*)d_in[1];
  const float* km = (const float*)d_in[2];
  float* out = (float*)d_out;

  dim3 grid(H_ / TILE_R, B_, COUT / CGRP);  // (8, 8, 4) = 256 blocks
  mnn_tropical_kernel<<<grid, BLOCK, 0, stream>>>(x, kh, km, out);

  (void)in_sizes; (void)n_in; (void)out_size; (void)d_ws; (void)ws_size;
}